// Corrosion_Refinement_15238543966313
// MI455X (gfx1250) — compile-verified
//
#include <hip/hip_runtime.h>
#include <stdint.h>

// ---------------- problem constants (from reference setup_inputs) ----------
#define NCURVE   2000          // curves per batch
#define NLINE    2000          // curve samples per batch (line_choice)
#define NSURF    4000          // surface points per batch
#define NFACE    8000          // faces per batch
#define NTRI_S   26000         // surface samples per batch (tri_choice)
#define NPTS     34000         // 2000 + 2000 + 4000 + 26000
#define OCC_N    258           // 256 + 2*2 pad - 3 + 1
#define OCC_PER_B ((size_t)OCC_N * OCC_N * OCC_N)   // 17,173,512 (divisible by 8)

typedef __attribute__((ext_vector_type(4))) float v4f;   // native vector for NT stores

// Dilation scatter: point (cx,cy,cz) -> occ[cx..cx+2, cy..cy+2, cz..cz+2] = 1.
// cube = clip(x*256+128.5, 0, 255) as int (trunc == floor, values >= 0).
__device__ __forceinline__ void scatter27(float* __restrict__ occ, int b,
                                          float x0, float x1, float x2) {
  int cx = (int)fminf(fmaxf(x0 * 256.0f + 128.5f, 0.0f), 255.0f);
  int cy = (int)fminf(fmaxf(x1 * 256.0f + 128.5f, 0.0f), 255.0f);
  int cz = (int)fminf(fmaxf(x2 * 256.0f + 128.5f, 0.0f), 255.0f);
  size_t base = (size_t)b * OCC_PER_B;
  for (int dx = 0; dx < 3; ++dx) {
    for (int dy = 0; dy < 3; ++dy) {
      size_t row = base + ((size_t)(cx + dx) * OCC_N + (size_t)(cy + dy)) * OCC_N + (size_t)cz;
      occ[row + 0] = 1.0f;
      occ[row + 1] = 1.0f;
      occ[row + 2] = 1.0f;
    }
  }
}

// ---------------------------------------------------------------------------
// Kernel 1: zero-fill the 275 MB occ region with non-temporal b128 stores.
// occ is write-once and larger than the 192 MB L2 -> stream past the caches.
// ---------------------------------------------------------------------------
__global__ void fill_occ(v4f* __restrict__ p, long n4) {
  long i      = (long)blockIdx.x * blockDim.x + threadIdx.x;
  long stride = (long)gridDim.x * blockDim.x;
  v4f z = (v4f)(0.0f);
  for (; i < n4; i += stride)
    __builtin_nontemporal_store(z, p + i);   // global_store_b128 th:NT
}

// ---------------------------------------------------------------------------
// Kernel 2: curves copy (0..1999), curve lerp samples (2000..3999),
//           surfaces copy (4000..7999). One thread per point.
// ---------------------------------------------------------------------------
__global__ void pts_base(const float* __restrict__ curves,
                         const float* __restrict__ surfaces,
                         const float* __restrict__ t_line,
                         const int*   __restrict__ lines_array,
                         const int*   __restrict__ line_choice,
                         float* __restrict__ x_out,
                         float* __restrict__ occ,
                         int B) {
  int idx = blockIdx.x * blockDim.x + threadIdx.x;
  if (idx >= B * (NCURVE + NLINE + NSURF)) return;
  int b     = idx / (NCURVE + NLINE + NSURF);
  int local = idx % (NCURVE + NLINE + NSURF);

  float x0, x1, x2;
  if (local < NCURVE) {
    const float* c = curves + ((size_t)b * NCURVE + local) * 3;
    x0 = c[0]; x1 = c[1]; x2 = c[2];
  } else if (local < NCURVE + NLINE) {
    int i   = local - NCURVE;
    int sel = line_choice[(size_t)b * NLINE + i];
    const int* ln = lines_array + ((size_t)b * NLINE + sel) * 2;
    const float* pa = curves + ((size_t)b * NCURVE + ln[0]) * 3;
    const float* pb = curves + ((size_t)b * NCURVE + ln[1]) * 3;
    float t = t_line[(size_t)b * NLINE + i];
    x0 = pa[0] + t * (pb[0] - pa[0]);
    x1 = pa[1] + t * (pb[1] - pa[1]);
    x2 = pa[2] + t * (pb[2] - pa[2]);
  } else {
    int i = local - (NCURVE + NLINE);
    const float* s = surfaces + ((size_t)b * NSURF + i) * 3;
    x0 = s[0]; x1 = s[1]; x2 = s[2];
  }

  size_t p = (size_t)b * NPTS + local;
  x_out[3 * p + 0] = x0;
  x_out[3 * p + 1] = x1;
  x_out[3 * p + 2] = x2;
  scatter27(occ, b, x0, x1, x2);
}

// ---------------------------------------------------------------------------
// Kernel 3: surface samples. Each block stages its batch's 48 KB surfaces
// table into LDS via CDNA5 async global->LDS loads (GVS mode), waits on
// ASYNCcnt, then gathers triangle vertices from LDS.
// ---------------------------------------------------------------------------
#define C_BLOCK        256
#define C_PTS_PER_BLK  1024

__global__ __launch_bounds__(C_BLOCK) void
pts_surface(const float* __restrict__ surfaces,
            const float* __restrict__ uv,
            const int*   __restrict__ faces_array,
            const int*   __restrict__ tri_choice,
            float* __restrict__ x_out,
            float* __restrict__ occ,
            int blocksPerBatch) {
  __shared__ float s_surf[NSURF * 3];          // 48,000 bytes

  int b     = blockIdx.x / blocksPerBatch;
  int chunk = blockIdx.x % blocksPerBatch;

  // LDS byte offset = low 32 bits of the flat pointer (LDS aperture mapping).
  // The ptrtoint escapes s_surf so the asm "memory" clobber pins its contents.
  uint32_t lds_base = (uint32_t)(uintptr_t)(void*)&s_surf[0];
  uint64_t gbase    = (uint64_t)(uintptr_t)(surfaces + (size_t)b * NSURF * 3);

  const int nchunks = (NSURF * 3 * 4) / 16;    // 3000 x 16B
  for (int c = threadIdx.x; c < nchunks; c += blockDim.x) {
    uint32_t off = (uint32_t)c * 16u;
    asm volatile("global_load_async_to_lds_b128 %0, %1, %2"
                 :: "v"(lds_base + off), "v"(off), "s"(gbase)
                 : "memory");
  }
  asm volatile("s_wait_asynccnt 0x0" ::: "memory");
  __syncthreads();

  int i0 = chunk * C_PTS_PER_BLK;
  for (int k = threadIdx.x; k < C_PTS_PER_BLK; k += blockDim.x) {
    int i = i0 + k;
    if (i >= NTRI_S) break;

    int tri = tri_choice[(size_t)b * NTRI_S + i];
    const int* f = faces_array + ((size_t)b * NFACE + tri) * 3;
    int ia = f[0], ib = f[1], ic = f[2];

    float u = uv[((size_t)b * NTRI_S + i) * 2 + 0];
    float v = uv[((size_t)b * NTRI_S + i) * 2 + 1];
    if (u + v > 1.0f) { u = 1.0f - u; v = 1.0f - v; }

    float tax = s_surf[ia * 3 + 0], tay = s_surf[ia * 3 + 1], taz = s_surf[ia * 3 + 2];
    float tbx = s_surf[ib * 3 + 0], tby = s_surf[ib * 3 + 1], tbz = s_surf[ib * 3 + 2];
    float tcx = s_surf[ic * 3 + 0], tcy = s_surf[ic * 3 + 1], tcz = s_surf[ic * 3 + 2];

    float x0 = tax + u * (tbx - tax) + v * (tcx - tax);
    float x1 = tay + u * (tby - tay) + v * (tcy - tay);
    float x2 = taz + u * (tbz - taz) + v * (tcz - taz);

    size_t p = (size_t)b * NPTS + (NCURVE + NLINE + NSURF) + i;
    x_out[3 * p + 0] = x0;
    x_out[3 * p + 1] = x1;
    x_out[3 * p + 2] = x2;
    scatter27(occ, b, x0, x1, x2);
  }
}

// ---------------------------------------------------------------------------
extern "C" void kernel_launch(void* const* d_in, const int* in_sizes, int n_in,
                              void* d_out, int out_size, void* d_ws, size_t ws_size,
                              hipStream_t stream) {
  (void)n_in; (void)out_size; (void)d_ws; (void)ws_size;
  // setup_inputs order:
  // 0 imgs (unused), 1 curves, 2 surfaces, 3 t_line, 4 uv,
  // 5 lines_array, 6 faces_array, 7 indices_array (unused),
  // 8 line_choice, 9 tri_choice
  const float* curves      = (const float*)d_in[1];
  const float* surfaces    = (const float*)d_in[2];
  const float* t_line      = (const float*)d_in[3];
  const float* uv          = (const float*)d_in[4];
  const int*   lines_array = (const int*)d_in[5];
  const int*   faces_array = (const int*)d_in[6];
  const int*   line_choice = (const int*)d_in[8];
  const int*   tri_choice  = (const int*)d_in[9];

  int B = in_sizes[1] / (NCURVE * 3);          // = 4

  float* x_out = (float*)d_out;                           // (B, 34000, 3)
  float* occ   = x_out + (size_t)B * NPTS * 3;            // (B, 1, 258^3)

  // 1) zero occ (OCC_PER_B divisible by 8 -> exact float4 count, 16B aligned)
  long n4 = (long)(((size_t)B * OCC_PER_B) / 4);
  fill_occ<<<16384, 256, 0, stream>>>((v4f*)occ, n4);

  // 2) curves / curve-samples / surfaces copy + scatter
  int totB = B * (NCURVE + NLINE + NSURF);
  pts_base<<<(totB + 255) / 256, 256, 0, stream>>>(
      curves, surfaces, t_line, lines_array, line_choice, x_out, occ, B);

  // 3) surface samples with async-LDS staging + scatter
  int bpb = (NTRI_S + C_PTS_PER_BLK - 1) / C_PTS_PER_BLK;  // 26
  pts_surface<<<B * bpb, C_BLOCK, 0, stream>>>(
      surfaces, uv, faces_array, tri_choice, x_out, occ, bpb);
}